// GNNModel_16827681865964
// MI455X (gfx1250) — compile-verified
//
#include <hip/hip_runtime.h>
#include <hip/hip_bf16.h>

typedef float v2f __attribute__((ext_vector_type(2)));
typedef float v8f __attribute__((ext_vector_type(8)));

#define D 128          // feature dim (D_IN == D_HID == 128)
#define TILE_M 16
#define A_STRIDE 132   // padded LDS stride to dodge bank conflicts

// ---------------------------------------------------------------------------
// 1) dis <- 1.0 (self-loop contribution to degree), out <- 0
// ---------------------------------------------------------------------------
__global__ __launch_bounds__(256) void init_kernel(float* __restrict__ dis,
                                                   float* __restrict__ out,
                                                   int n_nodes) {
    size_t idx = (size_t)blockIdx.x * 256 + threadIdx.x;
    size_t tot = (size_t)n_nodes * D;
    if (idx < (size_t)n_nodes) dis[idx] = 1.0f;
    if (idx < tot) out[idx] = 0.0f;
}

// ---------------------------------------------------------------------------
// 2) deg[dst[e]] += 1  (accumulated into the dis buffer)
// ---------------------------------------------------------------------------
__global__ __launch_bounds__(256) void degree_kernel(const long long* __restrict__ dst,
                                                     float* __restrict__ deg,
                                                     int n_edges) {
    int e = blockIdx.x * 256 + threadIdx.x;
    if (e < n_edges) {
        long long d = dst[e];
        __hip_atomic_fetch_add(&deg[d], 1.0f, __ATOMIC_RELAXED, __HIP_MEMORY_SCOPE_AGENT);
    }
}

// ---------------------------------------------------------------------------
// 3) dis <- deg > 0 ? rsqrt(max(deg,1e-12)) : 0
// ---------------------------------------------------------------------------
__global__ __launch_bounds__(256) void rsqrt_kernel(float* __restrict__ dis, int n) {
    int i = blockIdx.x * 256 + threadIdx.x;
    if (i < n) {
        float d = dis[i];
        dis[i] = (d > 0.0f) ? rsqrtf(fmaxf(d, 1e-12f)) : 0.0f;
    }
}

// ---------------------------------------------------------------------------
// 4) T = X @ W^T  via V_WMMA_F32_16X16X4_F32
//    Block = 256 threads (8 waves). Block handles 16 rows of X; wave w handles
//    output columns [16w, 16w+16). A tile staged in LDS; B (W rows) preloaded
//    into 64 VGPRs/wave; 32 chained WMMAs over K=128.
//    VGPR layouts per CDNA5 ISA 7.12.2:
//      A (16x4 f32):  lane m(0..15)/m+16 -> M=m; VGPR0=K{0,2}, VGPR1=K{1,3}
//      B (4x16 f32):  mirrored (K in lane-half+VGPR, N=lane%16)
//      C/D 16x16 f32: VGPR r, lanes 0-15 -> M=r, lanes 16-31 -> M=r+8; N=lane%16
// ---------------------------------------------------------------------------
__global__ __launch_bounds__(256) void gemm_xWT_kernel(const float* __restrict__ X,
                                                       const float* __restrict__ W,
                                                       float* __restrict__ T,
                                                       int nrows) {
    __shared__ float sA[TILE_M * A_STRIDE];

    const int tid  = threadIdx.x;
    const int row0 = blockIdx.x * TILE_M;

    // Cooperative load of the 16x128 A tile (float4 per thread iteration)
    for (int i = tid; i < TILE_M * (D / 4); i += 256) {
        int r  = i >> 5;      // row within tile (0..15), D/4 == 32
        int c4 = i & 31;      // float4 column
        int rr = row0 + r;
        if (rr >= nrows) rr = nrows - 1;           // clamp (N is 16-aligned anyway)
        float4 v = ((const float4*)(X + (size_t)rr * D))[c4];
        float* p = &sA[r * A_STRIDE + c4 * 4];
        p[0] = v.x; p[1] = v.y; p[2] = v.z; p[3] = v.w;
    }

    const int wave = tid >> 5;
    const int lane = tid & 31;
    const int m    = lane & 15;   // M for A / N for B / N for C
    const int hi   = lane >> 4;   // lane-half selects K offset (+2) / M offset (+8)
    const int n0   = wave * 16;

    // Preload B: wave needs W rows [n0, n0+16), all K. 32 * v2f = 64 VGPRs.
    v2f breg[32];
    const float* wrow = W + (size_t)(n0 + m) * D + 2 * hi;
#pragma unroll
    for (int s = 0; s < 32; ++s) {
        breg[s].x = wrow[4 * s];
        breg[s].y = wrow[4 * s + 1];
    }

    __syncthreads();

    v8f acc = {};
    const float* arow = &sA[m * A_STRIDE + 2 * hi];
#pragma unroll
    for (int s = 0; s < 32; ++s) {
        v2f a;
        a.x = arow[4 * s];
        a.y = arow[4 * s + 1];
        acc = __builtin_amdgcn_wmma_f32_16x16x4_f32(
            /*neg_a=*/false, a, /*neg_b=*/false, breg[s],
            /*c_mod=*/(short)0, acc, /*reuse_a=*/false, /*reuse_b=*/false);
    }

    // Store 16x16 f32 D tile
    float* outp = T + (size_t)row0 * D + n0 + m;
#pragma unroll
    for (int r = 0; r < 8; ++r) {
        int rr = r + 8 * hi;
        if (row0 + rr < nrows)
            outp[(size_t)rr * D] = acc[r];
    }
}

// ---------------------------------------------------------------------------
// 5) scatter: out[dst] += norm * T[src]; edges e>=E are self-loops (i,i)
//    1 wave per edge; each lane moves a float4 -> 4 global_atomic_add_f32
// ---------------------------------------------------------------------------
__global__ __launch_bounds__(256) void scatter_kernel(const float* __restrict__ T,
                                                      const long long* __restrict__ src,
                                                      const long long* __restrict__ dst,
                                                      const float* __restrict__ dis,
                                                      float* __restrict__ out,
                                                      int n_edges, int n_nodes) {
    int e    = blockIdx.x * 8 + (threadIdx.x >> 5);
    int lane = threadIdx.x & 31;
    int total = n_edges + n_nodes;
    if (e >= total) return;

    long long s, d;
    if (e < n_edges) { s = src[e]; d = dst[e]; }
    else             { s = d = (long long)(e - n_edges); }

    float norm = dis[s] * dis[d];

    float4 v = ((const float4*)(T + (size_t)s * D))[lane];
    float* o = out + (size_t)d * D + lane * 4;
    __hip_atomic_fetch_add(o + 0, v.x * norm, __ATOMIC_RELAXED, __HIP_MEMORY_SCOPE_AGENT);
    __hip_atomic_fetch_add(o + 1, v.y * norm, __ATOMIC_RELAXED, __HIP_MEMORY_SCOPE_AGENT);
    __hip_atomic_fetch_add(o + 2, v.z * norm, __ATOMIC_RELAXED, __HIP_MEMORY_SCOPE_AGENT);
    __hip_atomic_fetch_add(o + 3, v.w * norm, __ATOMIC_RELAXED, __HIP_MEMORY_SCOPE_AGENT);
}

// ---------------------------------------------------------------------------
// 6) h = relu(acc + b1); acc re-zeroed so d_out can be reused for layer 2
// ---------------------------------------------------------------------------
__global__ __launch_bounds__(256) void relu_bias_kernel(float* __restrict__ acc,
                                                        const float* __restrict__ b,
                                                        float* __restrict__ h,
                                                        size_t tot) {
    size_t idx = (size_t)blockIdx.x * 256 + threadIdx.x;
    if (idx < tot) {
        float v = acc[idx] + b[idx & (D - 1)];
        h[idx]  = v > 0.0f ? v : 0.0f;
        acc[idx] = 0.0f;
    }
}

// ---------------------------------------------------------------------------
// 7) out += b2
// ---------------------------------------------------------------------------
__global__ __launch_bounds__(256) void bias_kernel(float* __restrict__ out,
                                                   const float* __restrict__ b,
                                                   size_t tot) {
    size_t idx = (size_t)blockIdx.x * 256 + threadIdx.x;
    if (idx < tot) out[idx] += b[idx & (D - 1)];
}

// ---------------------------------------------------------------------------
extern "C" void kernel_launch(void* const* d_in, const int* in_sizes, int n_in,
                              void* d_out, int out_size, void* d_ws, size_t ws_size,
                              hipStream_t stream) {
    const float*      x  = (const float*)d_in[0];
    const long long*  ei = (const long long*)d_in[1];   // int64 [2, E]
    const float*      W1 = (const float*)d_in[2];
    const float*      b1 = (const float*)d_in[3];
    const float*      W2 = (const float*)d_in[4];
    const float*      b2 = (const float*)d_in[5];
    float*            out = (float*)d_out;

    const int N = in_sizes[0] / D;          // 100000
    const int E = in_sizes[1] / 2;          // 800000
    const long long* src = ei;
    const long long* dst = ei + E;

    // workspace layout: dis[N] | t[N*D] | h[N*D]
    float* dis = (float*)d_ws;
    size_t nAl = ((size_t)N + 255) & ~(size_t)255;
    float* t   = dis + nAl;
    float* h   = t + (size_t)N * D;

    const size_t tot = (size_t)N * D;
    const int blk = 256;
    const int gTot   = (int)((tot + blk - 1) / blk);
    const int gNode  = (N + blk - 1) / blk;
    const int gEdge  = (E + blk - 1) / blk;
    const int gGemm  = (N + TILE_M - 1) / TILE_M;
    const int gScat  = (E + N + 7) / 8;

    // degrees -> dis
    init_kernel<<<gTot, blk, 0, stream>>>(dis, out, N);
    degree_kernel<<<gEdge, blk, 0, stream>>>(dst, dis, E);
    rsqrt_kernel<<<gNode, blk, 0, stream>>>(dis, N);

    // layer 1: t = x @ W1^T ; out += scatter(norm * t[src]) ; h = relu(out+b1), out<-0
    gemm_xWT_kernel<<<gGemm, blk, 0, stream>>>(x, W1, t, N);
    scatter_kernel<<<gScat, blk, 0, stream>>>(t, src, dst, dis, out, E, N);
    relu_bias_kernel<<<gTot, blk, 0, stream>>>(out, b1, h, tot);

    // layer 2: t = h @ W2^T ; out += scatter(norm * t[src]) ; out += b2
    gemm_xWT_kernel<<<gGemm, blk, 0, stream>>>(h, W2, t, N);
    scatter_kernel<<<gScat, blk, 0, stream>>>(t, src, dst, dis, out, E, N);
    bias_kernel<<<gTot, blk, 0, stream>>>(out, b2, tot);
}